// CompositionalAgent_9431748182599
// MI455X (gfx1250) — compile-verified
//
#include <hip/hip_runtime.h>
#include <math.h>

#define H      2048
#define NGH    10240      // 5*H  (sigmoid rows)
#define TOT    12288      // 6*H
#define DICT   50000
#define FDIM   6
#define INW    16
#define HALF   1024
#define EPSF   1e-8f

// ---- d_out flat offsets (reference return order) ----
#define OFF_PI 0L
#define OFF_V  8L
#define OFF_H  9L
#define OFF_C  2057L
#define OFF_FK 4105L           // new_f_keys  50000*6
#define OFF_FV 304105L         // new_f_vals  50000*2048
#define OFF_RK 102704105L      // new_r_keys  50000
#define OFF_RV 102754105L      // new_r_vals  50000*2048

// ---- workspace offsets (floats) ----
#define WS_GATES 0             // 12288 activated preact (f,i,o,fun,rul,c_new)
#define WS_EF    12288         // exp(sim_f) [50000]
#define WS_ER    62288         // exp(sim_r) [50000]
#define WS_PSF   112288        // per-block exp partial sums (f) [256]
#define WS_PSR   112544
#define WS_SF    112800        // scalar sum_f
#define WS_SR    112801
#define WS_MEMF  112802        // tanh(dnd_read f) [2048]
#define WS_MEMR  114850
#define WS_HT    116898        // h_t [2048]
#define WS_HA    118946        // relu hidden [2048]
#define WS_PARTF 120996        // weighted-sum partials [100][2048] (16B aligned)
#define WS_PARTR 325796        // [100][2048]   (end = 530596 floats ~2.1 MB)

#define CHUNKS 100
#define ROWS_PER_CHUNK 500

typedef __attribute__((ext_vector_type(2))) float v2f;
typedef __attribute__((ext_vector_type(4))) float v4f;
typedef __attribute__((ext_vector_type(8))) float v8f;

// ---------------------------------------------------------------------------
// WMMA fp32 matvec core: wave computes y[rb..rb+15] = W[rb..rb+15,:] . vec
// A layout (16x4 f32, ISA 7.12.2): lanes 0-15 hold M=lane, K={k0,k0+1};
// lanes 16-31 hold M=lane-16, K={k0+2,k0+3}. B is the broadcast vector
// (all 16 columns identical), mirrored K-pairing across lane halves.
// Result: acc[v], lanes 0-15 -> y[rb+v]; lanes 16-31 -> y[rb+8+v].
// ---------------------------------------------------------------------------
__device__ __forceinline__ v8f wmma_matvec_row16(const float* __restrict__ W,
                                                 const float* __restrict__ vec,
                                                 int rb, int lane) {
  int half = lane >> 4;        // 0: K pair {0,1}, 1: K pair {2,3}
  int m    = lane & 15;        // row within tile
  const float* wr = W + (size_t)(rb + m) * H + 2 * half;
  const float* hv = vec + 2 * half;
  v8f acc = {0.f, 0.f, 0.f, 0.f, 0.f, 0.f, 0.f, 0.f};
#pragma unroll 8
  for (int k0 = 0; k0 < H; k0 += 4) {
    v2f a = *(const v2f*)(wr + k0);
    v2f b = *(const v2f*)(hv + k0);
    acc = __builtin_amdgcn_wmma_f32_16x16x4_f32(false, a, false, b,
                                                (short)0, acc, false, false);
  }
  return acc;
}

// K1: preact = x@W_i2h.T + h@W_h2h.T + biases; apply sigmoid/tanh -> ws.gates
__global__ __launch_bounds__(256) void k_gates(
    const float* __restrict__ x, const float* __restrict__ h,
    const float* __restrict__ W_i2h, const float* __restrict__ b_i2h,
    const float* __restrict__ W_h2h, const float* __restrict__ b_h2h,
    float* __restrict__ ws) {
  int lane = threadIdx.x & 31;
  int wave = (blockIdx.x * blockDim.x + threadIdx.x) >> 5;
  int rb = wave << 4;
  v8f acc = wmma_matvec_row16(W_h2h, h, rb, lane);
  if ((lane & 15) == 0) {                 // lanes 0 and 16 finalize 8 rows each
    int half = lane >> 4;
#pragma unroll
    for (int v = 0; v < 8; ++v) {
      int r = rb + half * 8 + v;
      float y = acc[v] + b_i2h[r] + b_h2h[r];
      const float* wi = W_i2h + r * INW;
#pragma unroll
      for (int k = 0; k < INW; ++k) y = fmaf(wi[k], x[k], y);
      float g = (r < NGH) ? 1.f / (1.f + expf(-y)) : tanhf(y);
      ws[WS_GATES + r] = g;
    }
  }
}

// K2: cosine sims -> exp(sim); per-block partial sums (deterministic tree)
__global__ __launch_bounds__(256) void k_sims(
    const float* __restrict__ x,
    const float* __restrict__ f_keys, const float* __restrict__ r_keys,
    float* __restrict__ ws) {
  __shared__ float sf[256], sr[256];
  int i = blockIdx.x * 256 + threadIdx.x;
  float qn2 = 0.f;
#pragma unroll
  for (int k = 0; k < FDIM; ++k) qn2 = fmaf(x[k], x[k], qn2);
  float nq  = sqrtf(qn2) + EPSF;
  float qr  = x[FDIM];
  float nqr = fabsf(qr) + EPSF;
  float ef = 0.f, er = 0.f;
  if (i < DICT) {
    const float* kr = f_keys + i * FDIM;
    float d = 0.f, n2 = 0.f;
#pragma unroll
    for (int k = 0; k < FDIM; ++k) { d = fmaf(kr[k], x[k], d); n2 = fmaf(kr[k], kr[k], n2); }
    ef = expf(d / ((sqrtf(n2) + EPSF) * nq));      // cosine in [-1,1]: exp safe
    ws[WS_EF + i] = ef;
    float rk = r_keys[i];
    er = expf((rk * qr) / ((fabsf(rk) + EPSF) * nqr));
    ws[WS_ER + i] = er;
  }
  sf[threadIdx.x] = ef; sr[threadIdx.x] = er;
  __syncthreads();
  for (int s = 128; s > 0; s >>= 1) {
    if (threadIdx.x < s) { sf[threadIdx.x] += sf[threadIdx.x + s];
                           sr[threadIdx.x] += sr[threadIdx.x + s]; }
    __syncthreads();
  }
  if (threadIdx.x == 0) { ws[WS_PSF + blockIdx.x] = sf[0];
                          ws[WS_PSR + blockIdx.x] = sr[0]; }
}

// K3: reduce per-block partials -> softmax denominators
__global__ __launch_bounds__(256) void k_reduce_sums(float* __restrict__ ws, int nblocks) {
  __shared__ float sf[256], sr[256];
  int t = threadIdx.x;
  sf[t] = (t < nblocks) ? ws[WS_PSF + t] : 0.f;
  sr[t] = (t < nblocks) ? ws[WS_PSR + t] : 0.f;
  __syncthreads();
  for (int s = 128; s > 0; s >>= 1) {
    if (t < s) { sf[t] += sf[t + s]; sr[t] += sr[t + s]; }
    __syncthreads();
  }
  if (t == 0) { ws[WS_SF] = sf[0]; ws[WS_SR] = sr[0]; }
}

// K4: fused dictionary copy + weighted-sum partials (the 1.6 GB hot loop).
// 400 blocks: [0..199] f_vals, [200..399] r_vals; block = (chunk, 1024-col tile).
// b128 non-temporal loads (aligned); stores stay scalar NT because OFF_FV/OFF_RV
// are == 1 (mod 4) floats, so 16B-aligned vector stores would be a lie.
__global__ __launch_bounds__(256) void k_dict(
    const float* __restrict__ f_vals, const float* __restrict__ r_vals,
    float* __restrict__ out, float* __restrict__ ws) {
  int b = blockIdx.x;
  int which = (b >= CHUNKS * 2) ? 1 : 0;
  int bb = which ? b - CHUNKS * 2 : b;
  int chunk = bb >> 1;
  int j4 = ((bb & 1) << 8) + threadIdx.x;       // float4 index 0..511
  size_t col = (size_t)j4 * 4;
  const float* vals = which ? r_vals : f_vals;
  float* outv = out + (which ? OFF_RV : OFF_FV);
  const float* e = ws + (which ? WS_ER : WS_EF);
  float* part = ws + (which ? WS_PARTR : WS_PARTF);
  v4f acc = {0.f, 0.f, 0.f, 0.f};
  int i0 = chunk * ROWS_PER_CHUNK;
#pragma unroll 4
  for (int i = i0; i < i0 + ROWS_PER_CHUNK; ++i) {
    size_t off = (size_t)i * H + col;
    v4f v = __builtin_nontemporal_load((const v4f*)(vals + off));   // b128
    float ei = e[i];
    acc += ei * v;
    __builtin_nontemporal_store(v.x, outv + off + 0);
    __builtin_nontemporal_store(v.y, outv + off + 1);
    __builtin_nontemporal_store(v.z, outv + off + 2);
    __builtin_nontemporal_store(v.w, outv + off + 3);
  }
  *(v4f*)(part + (size_t)chunk * H + col) = acc;   // WS_PART* are 16B aligned
}

// K_keys: copy key dicts with row write_idx replaced by the query
__global__ __launch_bounds__(256) void k_keys(
    const float* __restrict__ x,
    const float* __restrict__ f_keys, const float* __restrict__ r_keys,
    const int* __restrict__ widx_p, float* __restrict__ out) {
  int idx = blockIdx.x * 256 + threadIdx.x;
  int widx = *widx_p;
  if (idx < DICT * FDIM) {
    int row = idx / FDIM, col = idx - row * FDIM;
    out[OFF_FK + idx] = (row == widx) ? x[col] : f_keys[idx];
  } else {
    int i2 = idx - DICT * FDIM;
    if (i2 < DICT)
      out[OFF_RK + i2] = (i2 == widx) ? x[FDIM] : r_keys[i2];
  }
}

// K5: reduce chunk partials, normalize, tanh -> memfun/memrul
__global__ __launch_bounds__(256) void k_mem(float* __restrict__ ws) {
  int j = blockIdx.x * 256 + threadIdx.x;   // 0..2047
  float sf = 0.f, sr = 0.f;
#pragma unroll 4
  for (int c = 0; c < CHUNKS; ++c) {
    sf += ws[WS_PARTF + c * H + j];
    sr += ws[WS_PARTR + c * H + j];
  }
  ws[WS_MEMF + j] = tanhf(sf / ws[WS_SF]);
  ws[WS_MEMR + j] = tanhf(sr / ws[WS_SR]);
}

// K6: cell update, h_t, and the write_idx value rows (after K4's bulk copy)
__global__ __launch_bounds__(256) void k_cell(
    const float* __restrict__ c_in, const int* __restrict__ widx_p,
    float* __restrict__ out, float* __restrict__ ws) {
  int j = blockIdx.x * 256 + threadIdx.x;
  int widx = *widx_p;
  const float* g = ws + WS_GATES;
  float f   = g[j];
  float it  = g[H + j];
  float o   = g[2 * H + j];
  float fun = g[3 * H + j];
  float rul = g[4 * H + j];
  float cn  = g[5 * H + j];
  float ct = f * c_in[j] + it * cn + fun * ws[WS_MEMF + j] + rul * ws[WS_MEMR + j];
  float ht = o * tanhf(ct);
  out[OFF_C + j] = ct;
  out[OFF_H + j] = ht;
  ws[WS_HT + j] = ht;
  out[OFF_FV + (size_t)widx * H + j] = (j < HALF) ? ct : 0.f;   // c_ft
  out[OFF_RV + (size_t)widx * H + j] = (j < HALF) ? 0.f : ct;   // c_rt
}

// K7: ha = relu(W_ih @ h_t + b_ih) via WMMA matvec
__global__ __launch_bounds__(256) void k_hidden(
    const float* __restrict__ W_ih, const float* __restrict__ b_ih,
    float* __restrict__ ws) {
  int lane = threadIdx.x & 31;
  int wave = (blockIdx.x * blockDim.x + threadIdx.x) >> 5;
  int rb = wave << 4;
  v8f acc = wmma_matvec_row16(W_ih, ws + WS_HT, rb, lane);
  if ((lane & 15) == 0) {
    int half = lane >> 4;
#pragma unroll
    for (int v = 0; v < 8; ++v) {
      int r = rb + half * 8 + v;
      ws[WS_HA + r] = fmaxf(acc[v] + b_ih[r], 0.f);
    }
  }
}

// K8: actor/critic heads + softmax(8). 9 waves: waves 0-7 actor rows, wave 8 critic.
__global__ __launch_bounds__(288) void k_heads(
    const float* __restrict__ W_actor, const float* __restrict__ b_actor,
    const float* __restrict__ W_critic, const float* __restrict__ b_critic,
    float* __restrict__ out, const float* __restrict__ ws) {
  __shared__ float logits[8];
  int w = threadIdx.x >> 5, lane = threadIdx.x & 31;
  const float* ha = ws + WS_HA;
  const float* row = (w < 8) ? (W_actor + w * H) : W_critic;
  float s = 0.f;
  for (int k = lane; k < H; k += 32) s = fmaf(row[k], ha[k], s);
#pragma unroll
  for (int off = 16; off > 0; off >>= 1) s += __shfl_down(s, off);
  if (lane == 0) {
    if (w < 8) logits[w] = s + b_actor[w];
    else       out[OFF_V] = s + b_critic[0];
  }
  __syncthreads();
  if (threadIdx.x == 0) {
    float mx = logits[0];
#pragma unroll
    for (int i = 1; i < 8; ++i) mx = fmaxf(mx, logits[i]);
    float ex[8], sum = 0.f;
#pragma unroll
    for (int i = 0; i < 8; ++i) { ex[i] = expf(logits[i] - mx); sum += ex[i]; }
#pragma unroll
    for (int i = 0; i < 8; ++i) out[OFF_PI + i] = ex[i] / sum;
  }
}

extern "C" void kernel_launch(void* const* d_in, const int* in_sizes, int n_in,
                              void* d_out, int out_size, void* d_ws, size_t ws_size,
                              hipStream_t stream) {
  const float* x        = (const float*)d_in[0];
  const float* h        = (const float*)d_in[1];
  const float* c        = (const float*)d_in[2];
  const float* W_i2h    = (const float*)d_in[3];
  const float* b_i2h    = (const float*)d_in[4];
  const float* W_h2h    = (const float*)d_in[5];
  const float* b_h2h    = (const float*)d_in[6];
  const float* f_keys   = (const float*)d_in[7];
  const float* f_vals   = (const float*)d_in[8];
  const float* r_keys   = (const float*)d_in[9];
  const float* r_vals   = (const float*)d_in[10];
  const float* W_ih     = (const float*)d_in[11];
  const float* b_ih     = (const float*)d_in[12];
  const float* W_actor  = (const float*)d_in[13];
  const float* b_actor  = (const float*)d_in[14];
  const float* W_critic = (const float*)d_in[15];
  const float* b_critic = (const float*)d_in[16];
  const int*   widx     = (const int*)d_in[17];
  float* out = (float*)d_out;
  float* ws  = (float*)d_ws;

  k_gates      <<<96,   256, 0, stream>>>(x, h, W_i2h, b_i2h, W_h2h, b_h2h, ws);
  k_sims       <<<196,  256, 0, stream>>>(x, f_keys, r_keys, ws);
  k_reduce_sums<<<1,    256, 0, stream>>>(ws, 196);
  k_dict       <<<400,  256, 0, stream>>>(f_vals, r_vals, out, ws);
  k_keys       <<<1368, 256, 0, stream>>>(x, f_keys, r_keys, widx, out);
  k_mem        <<<8,    256, 0, stream>>>(ws);
  k_cell       <<<8,    256, 0, stream>>>(c, widx, out, ws);
  k_hidden     <<<16,   256, 0, stream>>>(W_ih, b_ih, ws);
  k_heads      <<<1,    288, 0, stream>>>(W_actor, b_actor, W_critic, b_critic, out, ws);
}